// SLALayer_10788957847932
// MI455X (gfx1250) — compile-verified
//
#include <hip/hip_runtime.h>

// ---------------------------------------------------------------------------
// SLA layer for MI455X (gfx1250, wave32, WMMA).
//
// Stage 1: qkv = x @ w_qkv^T (relu on q,k)   -- V_WMMA_F32_16X16X4_F32,
//          B staged in LDS (shared by 8 waves), A reused across 4 col tiles.
// Stage 2: kv[w,h,c,d] = seg-sum k*v ; s[w,h,c] = seg-sum k  -- VALU + LDS
// Stage 3: y = (q . kv) / (q . s + 1e-6)     -- VALU, coalesced kv reads
// Stage 4: out = y @ w_proj^T + b_proj       -- same WMMA scheme as stage 1
// ---------------------------------------------------------------------------

typedef __attribute__((ext_vector_type(2))) float v2f;
typedef __attribute__((ext_vector_type(4))) float v4f;
typedef __attribute__((ext_vector_type(8))) float v8f;

#define N_TOK   65536
#define C_DIM   128
#define H_HEADS 4
#define HD_DIM  32
#define NUM_WIN 1024
#define QKV_COLS 384            // 3 * C_DIM

// LDS row stride for staged B tiles: 132 floats.
//  - 132*4 = 528 bytes, multiple of 16  -> aligned b128 stores when filling
//  - 132 % 64 = 4 -> WMMA-pattern ds_load_b64 hits all 64 banks exactly once
#define BS_STRIDE 132

// ---------------------------------------------------------------------------
// Stage 1: qkv GEMM. Block = 8 waves; block owns 4 col tiles (64 cols of W,
// staged in LDS) x 8 row tiles (one per wave). K=128 in steps of 4.
// ---------------------------------------------------------------------------
__global__ __launch_bounds__(256)
void sla_qkv_gemm(const float* __restrict__ x,
                  const float* __restrict__ w_qkv,
                  float* __restrict__ qbuf,
                  float* __restrict__ kbuf,
                  float* __restrict__ vbuf)
{
    __shared__ float bs[64 * BS_STRIDE];

    const int ctg = blockIdx.x % (QKV_COLS / 64);   // 0..5: 64-col group
    const int rtg = blockIdx.x / (QKV_COLS / 64);   // 0..511: 8-row-tile group
    const int t   = threadIdx.x;

    // Cooperative load of B group: w_qkv rows [ctg*64, ctg*64+64) (64x128 f32).
#pragma unroll
    for (int j = 0; j < 8; ++j) {
        const int i  = t + j * 256;        // 0..2047 float4s
        const int r  = i >> 5;             // 0..63
        const int c4 = i & 31;             // 0..31
        v4f w4 = *(const v4f*)(w_qkv + (size_t)(ctg * 64 + r) * C_DIM + c4 * 4);
        *(v4f*)&bs[r * BS_STRIDE + c4 * 4] = w4;
    }
    __syncthreads();

    const int waveid = t >> 5;
    const int lane   = t & 31;
    const int nlo    = lane & 15;          // M for A, N for B/D
    const int half   = lane >> 4;          // half-wave selects K sub-pair
    const int koff   = half << 1;          // {0,2}
    const int rt     = rtg * 8 + waveid;

    const float* ap    = x + (size_t)(rt * 16 + nlo) * C_DIM + koff;
    const float* bbase = &bs[nlo * BS_STRIDE + koff];

    v8f acc[4];
#pragma unroll
    for (int cc = 0; cc < 4; ++cc)
        acc[cc] = (v8f){0.f, 0.f, 0.f, 0.f, 0.f, 0.f, 0.f, 0.f};

#pragma unroll
    for (int k = 0; k < C_DIM; k += 4) {
        v2f a = *(const v2f*)(ap + k);
#pragma unroll
        for (int cc = 0; cc < 4; ++cc) {
            v2f b = *(const v2f*)(bbase + cc * 16 * BS_STRIDE + k);
            acc[cc] = __builtin_amdgcn_wmma_f32_16x16x4_f32(
                false, a, false, b, (short)0, acc[cc], false, false);
        }
    }

    // Output select is uniform per block: 64-col group lies in one part.
    const int part = ctg >> 1;             // 0=q, 1=k, 2=v
    float* outb = (part == 0) ? qbuf : (part == 1) ? kbuf : vbuf;
    const bool do_relu = (part < 2);

#pragma unroll
    for (int cc = 0; cc < 4; ++cc) {
        const int n_g = (ctg * 4 + cc) * 16 + nlo;   // 0..383
        const int col = n_g & (C_DIM - 1);
#pragma unroll
        for (int r = 0; r < 8; ++r) {
            const int m_g = rt * 16 + r + (half << 3);
            float v = acc[cc][r];
            if (do_relu) v = fmaxf(v, 0.f);
            outb[(size_t)m_g * C_DIM + col] = v;
        }
    }
}

// ---------------------------------------------------------------------------
// Stage 2: per-window segment sums. One 256-thread block per window.
// Thread t owns (h = t>>6, c = (t>>1)&31, 16 consecutive d's selected by t&1).
// kv layout: (win, h, c, d)  -> stage-3 reads coalesce over d.
// ---------------------------------------------------------------------------
__global__ __launch_bounds__(256)
void sla_kv_segsum(const float* __restrict__ kbuf,
                   const float* __restrict__ vbuf,
                   const int*   __restrict__ offsets,
                   float* __restrict__ kvbuf,
                   float* __restrict__ sbuf)
{
    const int w     = blockIdx.x;
    const int start = offsets[w];
    const int end   = offsets[w + 1];
    const int t     = threadIdx.x;

    const int h  = t >> 6;
    const int c  = (t >> 1) & 31;
    const int d0 = (t & 1) << 4;      // 0 or 16

    __shared__ float sk[C_DIM];
    __shared__ float sv[C_DIM];

    float acc[16];
#pragma unroll
    for (int i = 0; i < 16; ++i) acc[i] = 0.f;
    float sacc = 0.f;

    for (int n = start; n < end; ++n) {
        __syncthreads();   // protect LDS from previous iteration's readers
        if (t < C_DIM)  sk[t]          = kbuf[(size_t)n * C_DIM + t];
        else            sv[t - C_DIM]  = vbuf[(size_t)n * C_DIM + (t - C_DIM)];
        __syncthreads();

        const float kval = sk[h * HD_DIM + c];
        const float* vrow = &sv[h * HD_DIM + d0];
#pragma unroll
        for (int i = 0; i < 16; ++i) acc[i] += kval * vrow[i];
        if (t < C_DIM) sacc += sk[t];
    }

    float* kvp = kvbuf + ((((size_t)w * H_HEADS + h) * HD_DIM + c) * HD_DIM + d0);
#pragma unroll
    for (int i = 0; i < 16; ++i) kvp[i] = acc[i];
    if (t < C_DIM) sbuf[(size_t)w * C_DIM + t] = sacc;
}

// ---------------------------------------------------------------------------
// Stage 3: y[n] = (q . kv[win]) / (q . s[win] + 1e-6).
// 256-thread block handles 2 tokens (128 threads each: h = tl>>5, d = tl&31).
// kv[win,h,c,d] reads are coalesced across d (=lane) for each c.
// ---------------------------------------------------------------------------
__global__ __launch_bounds__(256)
void sla_apply_kv(const float* __restrict__ qbuf,
                  const float* __restrict__ kvbuf,
                  const float* __restrict__ sbuf,
                  const int*   __restrict__ batch_win_inds,
                  float* __restrict__ ybuf)
{
    const int sub   = threadIdx.x >> 7;                  // 0 or 1
    const int tl    = threadIdx.x & 127;
    const int token = (blockIdx.x << 1) | sub;
    const int h     = tl >> 5;
    const int d     = tl & 31;

    __shared__ float qs[2][C_DIM];
    qs[sub][tl] = qbuf[(size_t)token * C_DIM + tl];
    __syncthreads();

    const int win = batch_win_inds[token];
    const float* qrow = &qs[sub][h * HD_DIM];
    const float* kvp  = kvbuf + (((size_t)win * H_HEADS + h) * HD_DIM) * HD_DIM + d;
    const float* sp   = sbuf + (size_t)win * C_DIM + h * HD_DIM;

    float y = 0.f, z = 0.f;
#pragma unroll
    for (int c = 0; c < HD_DIM; ++c) {
        const float qc = qrow[c];
        y += qc * kvp[(size_t)c * HD_DIM];
        z += qc * sp[c];
    }
    ybuf[(size_t)token * C_DIM + tl] = y / (z + 1e-6f);
}

// ---------------------------------------------------------------------------
// Stage 4: out = y @ w_proj^T + b_proj. Same LDS-staged WMMA scheme,
// 2 col groups of 4 tiles, accumulators pre-seeded with the bias.
// ---------------------------------------------------------------------------
__global__ __launch_bounds__(256)
void sla_proj_gemm(const float* __restrict__ ybuf,
                   const float* __restrict__ w_proj,
                   const float* __restrict__ b_proj,
                   float* __restrict__ out)
{
    __shared__ float bs[64 * BS_STRIDE];

    const int ctg = blockIdx.x % (C_DIM / 64);   // 0..1
    const int rtg = blockIdx.x / (C_DIM / 64);   // 0..511
    const int t   = threadIdx.x;

#pragma unroll
    for (int j = 0; j < 8; ++j) {
        const int i  = t + j * 256;
        const int r  = i >> 5;
        const int c4 = i & 31;
        v4f w4 = *(const v4f*)(w_proj + (size_t)(ctg * 64 + r) * C_DIM + c4 * 4);
        *(v4f*)&bs[r * BS_STRIDE + c4 * 4] = w4;
    }
    __syncthreads();

    const int waveid = t >> 5;
    const int lane   = t & 31;
    const int nlo    = lane & 15;
    const int half   = lane >> 4;
    const int koff   = half << 1;
    const int rt     = rtg * 8 + waveid;

    const float* ap    = ybuf + (size_t)(rt * 16 + nlo) * C_DIM + koff;
    const float* bbase = &bs[nlo * BS_STRIDE + koff];

    v8f acc[4];
#pragma unroll
    for (int cc = 0; cc < 4; ++cc) {
        const float bias = b_proj[ctg * 64 + cc * 16 + nlo];
        acc[cc] = (v8f){bias, bias, bias, bias, bias, bias, bias, bias};
    }

#pragma unroll
    for (int k = 0; k < C_DIM; k += 4) {
        v2f a = *(const v2f*)(ap + k);
#pragma unroll
        for (int cc = 0; cc < 4; ++cc) {
            v2f b = *(const v2f*)(bbase + cc * 16 * BS_STRIDE + k);
            acc[cc] = __builtin_amdgcn_wmma_f32_16x16x4_f32(
                false, a, false, b, (short)0, acc[cc], false, false);
        }
    }

#pragma unroll
    for (int cc = 0; cc < 4; ++cc) {
        const int col = ctg * 64 + cc * 16 + nlo;
#pragma unroll
        for (int r = 0; r < 8; ++r) {
            const int m_g = rt * 16 + r + (half << 3);
            out[(size_t)m_g * C_DIM + col] = acc[cc][r];
        }
    }
}

// ---------------------------------------------------------------------------
// Launch
// ---------------------------------------------------------------------------
extern "C" void kernel_launch(void* const* d_in, const int* in_sizes, int n_in,
                              void* d_out, int out_size, void* d_ws, size_t ws_size,
                              hipStream_t stream)
{
    const float* x              = (const float*)d_in[0];
    const float* w_qkv          = (const float*)d_in[1];
    const float* w_proj         = (const float*)d_in[2];
    const float* b_proj         = (const float*)d_in[3];
    const int*   batch_win_inds = (const int*)d_in[4];
    const int*   offsets        = (const int*)d_in[5];
    // d_in[6] = counts (unused; offsets suffice)

    float* ws   = (float*)d_ws;
    float* qbuf = ws;                                   // N*128
    float* kbuf = qbuf + (size_t)N_TOK * C_DIM;         // N*128
    float* vbuf = kbuf + (size_t)N_TOK * C_DIM;         // N*128
    float* ybuf = vbuf + (size_t)N_TOK * C_DIM;         // N*128
    float* kvbuf = ybuf + (size_t)N_TOK * C_DIM;        // NUM_WIN*4*32*32
    float* sbuf  = kvbuf + (size_t)NUM_WIN * H_HEADS * HD_DIM * HD_DIM; // NUM_WIN*128

    float* out = (float*)d_out;

    // Stage 1: 512 row-tile groups * 6 col groups = 3072 blocks.
    sla_qkv_gemm<<<(N_TOK / 128) * (QKV_COLS / 64), 256, 0, stream>>>(
        x, w_qkv, qbuf, kbuf, vbuf);

    // Stage 2: one block per window.
    sla_kv_segsum<<<NUM_WIN, 256, 0, stream>>>(kbuf, vbuf, offsets, kvbuf, sbuf);

    // Stage 3: 2 tokens per block.
    sla_apply_kv<<<N_TOK / 2, 256, 0, stream>>>(qbuf, kvbuf, sbuf, batch_win_inds, ybuf);

    // Stage 4: 512 row-tile groups * 2 col groups = 1024 blocks.
    sla_proj_gemm<<<(N_TOK / 128) * (C_DIM / 64), 256, 0, stream>>>(
        ybuf, w_proj, b_proj, out);
}